// RightPool_1417339208204
// MI455X (gfx1250) — compile-verified
//
#include <hip/hip_runtime.h>

// RightPool / cummax along width (last dim, W=128) for an (8,256,128,128) f32 tensor.
// Pure streaming kernel (~268 MB traffic, ~11.5 us floor at 23.3 TB/s).
// Fast path: one wave32 owns 4 consecutive rows, NO bounds logic (grid covers
// exactly the full tiles). 4 independent NT b128 loads issued back-to-back for
// MLP, clamped Hillis-Steele max-scan via ds_bpermute (wave32 lane-permute HW,
// 5 steps), NT b128 stores. A tiny guarded tail kernel covers n_rows % 32 != 0
// (never taken for the reference shape; host branch is deterministic).

typedef __attribute__((ext_vector_type(4))) float v4f;

#define ROW_W 128            // floats per row (last dim)
#define WAVES_PER_BLOCK 8    // 256 threads = 8 wave32
#define ROWS_PER_WAVE 4
#define ROWS_PER_BLOCK (WAVES_PER_BLOCK * ROWS_PER_WAVE)  // 32

__device__ __forceinline__ float bperm_f32(int byte_idx, float v) {
  return __int_as_float(__builtin_amdgcn_ds_bpermute(byte_idx, __float_as_int(v)));
}

// Shared per-row body: in-lane cummax + wave32 max-scan + combine.
__device__ __forceinline__ v4f row_cummax(v4f v, int lane, const int idx[5],
                                          int idx_prev) {
  float m0 = v.x;
  float m1 = fmaxf(m0, v.y);
  float m2 = fmaxf(m1, v.z);
  float m3 = fmaxf(m2, v.w);

  // Clamped Hillis-Steele inclusive max-scan (idempotent op => no predication).
  float agg = m3;
#pragma unroll
  for (int i = 0; i < 5; ++i)
    agg = fmaxf(agg, bperm_f32(idx[i], agg));

  // Exclusive prefix = inclusive of lane-1; lane 0 -> -inf.
  float pre = bperm_f32(idx_prev, agg);
  pre = (lane > 0) ? pre : -__builtin_inff();

  v4f o;
  o.x = fmaxf(pre, m0);
  o.y = fmaxf(pre, m1);
  o.z = fmaxf(pre, m2);
  o.w = fmaxf(pre, m3);
  return o;
}

__device__ __forceinline__ void scan_indices(int lane, int idx[5], int& idx_prev) {
#pragma unroll
  for (int i = 0, d = 1; d < 32; d <<= 1, ++i)
    idx[i] = ((lane >= d) ? (lane - d) : 0) << 2;
  idx_prev = ((lane > 0) ? (lane - 1) : 0) << 2;
}

// Fast path: grid covers exactly (n_rows / 32) blocks; zero bounds checks.
__global__ __launch_bounds__(256) void rightpool_full_kernel(
    const float* __restrict__ x, float* __restrict__ out) {
  const int lane = threadIdx.x & 31;
  const int wave = threadIdx.x >> 5;
  const unsigned row0 = blockIdx.x * ROWS_PER_BLOCK + wave * ROWS_PER_WAVE;

  int idx[5], idx_prev;
  scan_indices(lane, idx, idx_prev);

  const v4f* __restrict__ xin =
      reinterpret_cast<const v4f*>(x + (size_t)row0 * ROW_W) + lane;
  v4f* __restrict__ oout =
      reinterpret_cast<v4f*>(out + (size_t)row0 * ROW_W) + lane;

  // 4 independent in-flight NT b128 loads (512 floats contiguous per wave).
  v4f v[ROWS_PER_WAVE];
#pragma unroll
  for (int r = 0; r < ROWS_PER_WAVE; ++r)
    v[r] = __builtin_nontemporal_load(xin + (size_t)r * (ROW_W / 4));

#pragma unroll
  for (int r = 0; r < ROWS_PER_WAVE; ++r) {
    v4f o = row_cummax(v[r], lane, idx, idx_prev);
    __builtin_nontemporal_store(o, oout + (size_t)r * (ROW_W / 4));
  }
}

// Generic tail: one row per wave, guarded. (Unused for the reference shape.)
__global__ __launch_bounds__(256) void rightpool_tail_kernel(
    const float* __restrict__ x, float* __restrict__ out,
    int row_base, int n_rows) {
  const int lane = threadIdx.x & 31;
  const int wave = threadIdx.x >> 5;
  const int row = row_base + blockIdx.x * WAVES_PER_BLOCK + wave;
  if (row >= n_rows) return;  // uniform per wave

  int idx[5], idx_prev;
  scan_indices(lane, idx, idx_prev);

  const v4f* __restrict__ xin =
      reinterpret_cast<const v4f*>(x + (size_t)row * ROW_W);
  v4f v = __builtin_nontemporal_load(xin + lane);
  v4f o = row_cummax(v, lane, idx, idx_prev);
  v4f* __restrict__ oout = reinterpret_cast<v4f*>(out + (size_t)row * ROW_W);
  __builtin_nontemporal_store(o, oout + lane);
}

extern "C" void kernel_launch(void* const* d_in, const int* in_sizes, int n_in,
                              void* d_out, int out_size, void* d_ws, size_t ws_size,
                              hipStream_t stream) {
  (void)n_in; (void)d_ws; (void)ws_size; (void)in_sizes;
  const float* x = (const float*)d_in[0];
  float* out     = (float*)d_out;

  const int n_rows     = out_size / ROW_W;              // 262144 for ref shape
  const int full_blocks = n_rows / ROWS_PER_BLOCK;      // 8192
  const int full_rows   = full_blocks * ROWS_PER_BLOCK;
  const int tail_rows   = n_rows - full_rows;           // 0 for ref shape

  if (full_blocks > 0)
    rightpool_full_kernel<<<full_blocks, 256, 0, stream>>>(x, out);

  if (tail_rows > 0) {
    const int tail_blocks = (tail_rows + WAVES_PER_BLOCK - 1) / WAVES_PER_BLOCK;
    rightpool_tail_kernel<<<tail_blocks, 256, 0, stream>>>(x, out, full_rows,
                                                           n_rows);
  }
}